// WeightedAttention_89026082111903
// MI455X (gfx1250) — compile-verified
//
#include <hip/hip_runtime.h>

typedef __attribute__((ext_vector_type(2))) float v2f;
typedef __attribute__((ext_vector_type(4))) float v4f;
typedef __attribute__((ext_vector_type(8))) float v8f;

#define NEG_INF (-1e30f)

constexpr int Bn     = 32;
constexpr int Tn     = 4096;
constexpr int Dn     = 256;
constexpr int CHUNKS = 32;            // T-chunks per batch row -> 1024 blocks
constexpr int TCHUNK = Tn / CHUNKS;   // 128 timesteps per block
constexpr int TILE   = 16;            // timesteps per wave (WMMA M)
constexpr int WAVES  = TCHUNK / TILE; // 8 waves / block
constexpr int RS     = 260;           // padded LDS row stride (floats), conflict-free

// ---------------- Pass 1: per-(batch, chunk) split-softmax partials ----------------
__global__ __launch_bounds__(256) void wa_pass1(
    const float* __restrict__ seq, const long long* __restrict__ lengths,
    const float* __restrict__ att,
    float* __restrict__ pm, float* __restrict__ ps, float* __restrict__ pacc)
{
  extern __shared__ float smem[];
  float* sAtt  = smem;                       // 256
  float* sTile = sAtt + 256;                 // WAVES * TILE * RS
  float* sRm   = sTile + WAVES * TILE * RS;  // WAVES
  float* sRs   = sRm + WAVES;                // WAVES
  float* sRacc = sRs + WAVES;                // WAVES * 256

  const int tid   = threadIdx.x;
  const int lane  = tid & 31;
  const int wv    = tid >> 5;
  const int bidx  = blockIdx.x;
  const int b     = bidx / CHUNKS;
  const int chunk = bidx % CHUNKS;
  const int t0    = chunk * TCHUNK + wv * TILE;
  const long long len = lengths[b];

  sAtt[tid] = att[tid];

  // ---- stage this wave's 16 x 256 tile of seq into LDS (coalesced, streaming) ----
  const float* gsrc = seq + ((size_t)b * Tn + t0) * Dn;
  float* tile = sTile + wv * TILE * RS;
  __builtin_prefetch(gsrc + TILE * Dn + lane * 32, 0, 0);  // hint next tile
#pragma unroll
  for (int it = 0; it < (TILE * Dn) / (32 * 4); ++it) {    // 32 x float4 per lane
    const int idx = it * 128 + lane * 4;                   // 0..4095
    const int r = idx >> 8, c = idx & (Dn - 1);
    const v4f v = __builtin_nontemporal_load((const v4f*)(gsrc + r * Dn + c));
    *(v4f*)&tile[r * RS + c] = v;
  }
  __syncthreads();  // sAtt visible; also orders the in-wave LDS writes

  // ---- 16 logits via chained V_WMMA_F32_16X16X4_F32 (exact fp32) ----
  // A (16x4 f32): lanes 0-15 -> M=lane, K=k0+{0,1}; lanes 16-31 -> M=lane-16, K=k0+{2,3}
  // B (4x16): value depends only on K (att broadcast over N) -> layout-robust
  const int row  = lane & 15;
  const int koff = (lane >> 4) << 1;  // 0 or 2
  v8f c8 = {};
#pragma unroll
  for (int k0 = 0; k0 < Dn; k0 += 4) {
    v2f a = *(const v2f*)&tile[row * RS + k0 + koff];
    v2f bb;
    bb.x = sAtt[k0 + koff];
    bb.y = sAtt[k0 + koff + 1];
    c8 = __builtin_amdgcn_wmma_f32_16x16x4_f32(false, a, false, bb,
                                               (short)0, c8, false, false);
  }

  // D layout: lanes 0-15 hold rows 0..7 in c8[0..7]; lanes 16-31 hold rows 8..15.
  // Swap halves so every lane has all 16 logits.
  float lg[16];
#pragma unroll
  for (int r = 0; r < 8; ++r) {
    const float own = c8[r];
    const float oth = __shfl_xor(own, 16, 32);
    lg[r]     = (lane < 16) ? own : oth;
    lg[8 + r] = (lane < 16) ? oth : own;
  }

  // ---- masked local softmax partial over this wave's 16 timesteps ----
  float m = NEG_INF;
#pragma unroll
  for (int t = 0; t < 16; ++t)
    if ((long long)(t0 + t) < len) m = fmaxf(m, lg[t]);

  float s = 0.f;
  float wgt[16];
#pragma unroll
  for (int t = 0; t < 16; ++t) {
    const float e = ((long long)(t0 + t) < len) ? __expf(lg[t] - m) : 0.f;
    wgt[t] = e;
    s += e;
  }

  // acc over d: lane covers d = lane + 32*j  (conflict-free LDS reads)
  float acc[8] = {0.f, 0.f, 0.f, 0.f, 0.f, 0.f, 0.f, 0.f};
#pragma unroll
  for (int t = 0; t < 16; ++t) {
    const float wt = wgt[t];
#pragma unroll
    for (int j = 0; j < 8; ++j)
      acc[j] += wt * tile[t * RS + lane + 32 * j];
  }

  // ---- combine the 8 wave partials (log-sum-exp merge) ----
  if (lane == 0) { sRm[wv] = m; sRs[wv] = s; }
#pragma unroll
  for (int j = 0; j < 8; ++j) sRacc[wv * 256 + lane + 32 * j] = acc[j];
  __syncthreads();

  float M = NEG_INF;
#pragma unroll
  for (int v = 0; v < WAVES; ++v) M = fmaxf(M, sRm[v]);
  float S = 0.f, a = 0.f;
#pragma unroll
  for (int v = 0; v < WAVES; ++v) {
    const float e = __expf(sRm[v] - M);  // empty waves: s=0, acc=0 -> no effect
    S += sRs[v] * e;
    a += sRacc[v * 256 + tid] * e;
  }

  pacc[(size_t)bidx * Dn + tid] = a;
  if (tid == 0) { pm[bidx] = M; ps[bidx] = S; }
}

// ---------------- Pass 2: merge chunk partials per batch row ----------------
__global__ __launch_bounds__(256) void wa_pass2(
    const float* __restrict__ pm, const float* __restrict__ ps,
    const float* __restrict__ pacc, float* __restrict__ out)
{
  const int b = blockIdx.x, tid = threadIdx.x;
  float M = NEG_INF;
#pragma unroll
  for (int c = 0; c < CHUNKS; ++c) M = fmaxf(M, pm[b * CHUNKS + c]);
  float S = 0.f, a = 0.f;
#pragma unroll
  for (int c = 0; c < CHUNKS; ++c) {
    const float e = __expf(pm[b * CHUNKS + c] - M);
    S += ps[b * CHUNKS + c] * e;
    a += pacc[((size_t)(b * CHUNKS + c)) * Dn + tid] * e;
  }
  out[b * Dn + tid] = a / S;  // len >= 1 guarantees S > 0
}

extern "C" void kernel_launch(void* const* d_in, const int* in_sizes, int n_in,
                              void* d_out, int out_size, void* d_ws, size_t ws_size,
                              hipStream_t stream) {
  const float*     seq     = (const float*)d_in[0];
  const long long* lengths = (const long long*)d_in[1];
  const float*     att     = (const float*)d_in[2];

  // workspace: pm[1024] | ps[1024] | pacc[1024*256]  (~1.01 MB)
  float* pm   = (float*)d_ws;
  float* ps   = pm + Bn * CHUNKS;
  float* pacc = ps + Bn * CHUNKS;

  const size_t smem_bytes =
      (256 + (size_t)WAVES * TILE * RS + WAVES + WAVES + WAVES * 256) * sizeof(float);

  wa_pass1<<<Bn * CHUNKS, 256, smem_bytes, stream>>>(seq, lengths, att, pm, ps, pacc);
  wa_pass2<<<Bn, 256, 0, stream>>>(pm, ps, pacc, (float*)d_out);
}